// PIGNN_55130200211493
// MI455X (gfx1250) — compile-verified
//
#include <hip/hip_runtime.h>
#include <hip/hip_bf16.h>

typedef __attribute__((ext_vector_type(2))) float v2f;
typedef __attribute__((ext_vector_type(8))) float v8f;

// ---------------------------------------------------------------- init / cnt
__global__ void pignn_init_cnt(float* cnt, int n, float* stab, float* opf) {
    int i = blockIdx.x * blockDim.x + threadIdx.x;
    if (i < n) cnt[i] = 1.0f;                 // self-loop contributes 1 edge
    if (i == 0) { *stab = 0.0f; *opf = 0.0f; }
}

__global__ void pignn_count(const long long* __restrict__ ei, int E, float* cnt) {
    int e = blockIdx.x * blockDim.x + threadIdx.x;
    if (e < E) {
        int d = (int)ei[(long long)E + e];
        unsafeAtomicAdd(&cnt[d], 1.0f);
    }
}

__global__ void pignn_recip(float* cnt, int n) {
    int i = blockIdx.x * blockDim.x + threadIdx.x;
    if (i < n) cnt[i] = 1.0f / cnt[i];        // becomes inv_cnt
}

__global__ void pignn_copy(float* __restrict__ dst, const float* __restrict__ src, int n) {
    int i = blockIdx.x * blockDim.x + threadIdx.x;
    if (i < n) dst[i] = src[i];
}

// pad [N,5] row-major into [N,8] with zero cols 5..7 (also used as self-loop init)
__global__ void pignn_pad5to8(float* __restrict__ dst, const float* __restrict__ src, int n) {
    int i = blockIdx.x * blockDim.x + threadIdx.x;   // over n*8
    if (i < n * 8) {
        int r = i >> 3, f = i & 7;
        dst[i] = (f < 5) ? src[r * 5 + f] : 0.0f;
    }
}

// ------------------------------------------------------------- edge scatters
// layer-1 aggregate: read 5-wide x, accumulate into stride-8 padded xbar
__global__ void pignn_scatter5(const long long* __restrict__ ei, int E,
                               const float* __restrict__ X, float* xb) {
    int e = blockIdx.x * blockDim.x + threadIdx.x;
    if (e < E) {
        int s = (int)ei[e];
        int d = (int)ei[(long long)E + e];
        const float* xs = X + s * 5;
        float* xd = xb + d * 8;
#pragma unroll
        for (int f = 0; f < 5; ++f) unsafeAtomicAdd(&xd[f], xs[f]);
    }
}

__global__ void pignn_scatter64(const long long* __restrict__ ei, int E,
                                const float* __restrict__ X, float* xb) {
    int t = blockIdx.x * blockDim.x + threadIdx.x;
    int e = t >> 3;                            // 8 threads per edge
    int c = (t & 7) * 8;                       // 8 floats per thread
    if (e < E) {
        int s = (int)ei[e];
        int d = (int)ei[(long long)E + e];
        const float* xs = X + s * 64 + c;
        float* xd = xb + d * 64 + c;
#pragma unroll
        for (int f = 0; f < 8; ++f) unsafeAtomicAdd(&xd[f], xs[f]);
    }
}

// ----------------------------------------------- fold msg-linear into update
// T[0:Dpad,64]: rows<Din copy of U[:Din,:], rows>=Din zero
// M[0:Dpad,64]: rows<Din = W @ U[Din:,:], rows>=Din zero
// dv = b @ U[Din:,:] + c
__global__ void pignn_fuse_w(const float* __restrict__ W, const float* __restrict__ b,
                             const float* __restrict__ U, const float* __restrict__ c,
                             int Din, int Dpad, float* T, float* M, float* dv) {
    int tid = blockIdx.x * blockDim.x + threadIdx.x;
    int stride = gridDim.x * blockDim.x;
    int total = Dpad * 64;
    for (int i = tid; i < total; i += stride) {
        int r = i >> 6;
        T[i] = (r < Din) ? U[i] : 0.0f;
    }
    for (int i = tid; i < total; i += stride) {
        int r = i >> 6, n = i & 63;
        float s = 0.0f;
        if (r < Din)
            for (int k = 0; k < 64; ++k) s += W[r * 64 + k] * U[(Din + k) * 64 + n];
        M[i] = s;
    }
    for (int i = tid; i < 64; i += stride) {
        float s = c[i];
        for (int k = 0; k < 64; ++k) s += b[k] * U[(Din + k) * 64 + i];
        dv[i] = s;
    }
}

// ---------------------------------------------------------- WMMA layer GEMM
// H[n, 0:64] = relu( X[n]@T + inv_cnt[n]*XB[n]@M + dv ),  DIN % 4 == 0
// One wave32 computes a 16x64 tile via V_WMMA_F32_16X16X4_F32.
template <int DIN>
__global__ __launch_bounds__(256) void pignn_gemm(
    const float* X, const float* XB, const float* __restrict__ icnt,
    const float* __restrict__ T, const float* __restrict__ M,
    const float* __restrict__ dv, float* H, int Nn) {
    int wave = threadIdx.x >> 5;
    int lane = threadIdx.x & 31;
    int tile = blockIdx.x * 8 + wave;
    int row0 = tile * 16;
    if (row0 >= Nn) return;                    // uniform per wave: EXEC stays full

    int half = lane >> 4;                      // 0: K0/K1, 1: K2/K3 (ISA A/B layout)
    int lrow = lane & 15;
    int row  = row0 + lrow;
    float ic = icnt[row];

    v8f z = {};
    v8f acc[4] = {z, z, z, z};

#pragma unroll
    for (int pass = 0; pass < 2; ++pass) {
        const float* Ax = pass ? XB : X;
        const float* Bw = pass ? M : T;
        float scale = pass ? ic : 1.0f;
#pragma unroll
        for (int k0 = 0; k0 < DIN; k0 += 4) {
            int ka = k0 + half * 2;
            v2f a;
            a.x = Ax[row * DIN + ka] * scale;
            a.y = Ax[row * DIN + ka + 1] * scale;
#pragma unroll
            for (int j = 0; j < 4; ++j) {
                int n = j * 16 + lrow;
                v2f bb;
                bb.x = Bw[ka * 64 + n];
                bb.y = Bw[(ka + 1) * 64 + n];
                acc[j] = __builtin_amdgcn_wmma_f32_16x16x4_f32(
                    false, a, false, bb, (short)0, acc[j], false, false);
            }
        }
    }

    // C/D layout: lane half 0 -> rows 0..7, half 1 -> rows 8..15; col = j*16+lrow
#pragma unroll
    for (int j = 0; j < 4; ++j) {
        int n = j * 16 + lrow;
        float bias = dv[n];
#pragma unroll
        for (int v = 0; v < 8; ++v) {
            int m = v + 8 * half;
            float val = acc[j][v] + bias;
            H[(row0 + m) * 64 + n] = val > 0.0f ? val : 0.0f;
        }
    }
}

// ------------------------------------------------------------------- heads
__global__ __launch_bounds__(256) void pignn_heads(
    const float* __restrict__ H,
    const float* __restrict__ Ws, const float* __restrict__ bs,
    const float* __restrict__ Wst, const float* __restrict__ bst,
    const float* __restrict__ Wo, const float* __restrict__ bo,
    float* state, float* stab, float* opf, int Nn) {
    __shared__ float sWs[128], sWst[64], sWo[64], red[256];
    int tid = threadIdx.x;
    if (tid < 128) sWs[tid] = Ws[tid];
    if (tid < 64) { sWst[tid] = Wst[tid]; sWo[tid] = Wo[tid]; }
    __syncthreads();

    int n = blockIdx.x * 256 + tid;
    float st = 0.0f, so = 0.0f;
    if (n < Nn) {
        float s0 = 0.0f, s1 = 0.0f;
        const float* h = H + n * 64;
        for (int k = 0; k < 64; ++k) {
            float v = h[k];
            s0 += v * sWs[k * 2 + 0];
            s1 += v * sWs[k * 2 + 1];
            st += v * sWst[k];
            so += v * sWo[k];
        }
        state[n * 2 + 0] = s0 + bs[0];
        state[n * 2 + 1] = s1 + bs[1];
        st = 1.0f / (1.0f + __expf(-(st + bst[0])));
        so = so + bo[0];
    }
    float invN = 1.0f / (float)Nn;
    red[tid] = st; __syncthreads();
    for (int s = 128; s > 0; s >>= 1) { if (tid < s) red[tid] += red[tid + s]; __syncthreads(); }
    if (tid == 0) unsafeAtomicAdd(stab, red[0] * invN);
    __syncthreads();
    red[tid] = so; __syncthreads();
    for (int s = 128; s > 0; s >>= 1) { if (tid < s) red[tid] += red[tid + s]; __syncthreads(); }
    if (tid == 0) unsafeAtomicAdd(opf, red[0] * invN);
}

// ------------------------------------------------------------------ launch
extern "C" void kernel_launch(void* const* d_in, const int* in_sizes, int n_in,
                              void* d_out, int out_size, void* d_ws, size_t ws_size,
                              hipStream_t stream) {
    const float* x   = (const float*)d_in[0];
    const long long* ei = (const long long*)d_in[1];
    const float *W1 = (const float*)d_in[2],  *b1 = (const float*)d_in[3];
    const float *U1 = (const float*)d_in[4],  *c1 = (const float*)d_in[5];
    const float *W2 = (const float*)d_in[6],  *b2 = (const float*)d_in[7];
    const float *U2 = (const float*)d_in[8],  *c2 = (const float*)d_in[9];
    const float *W3 = (const float*)d_in[10], *b3 = (const float*)d_in[11];
    const float *U3 = (const float*)d_in[12], *c3 = (const float*)d_in[13];
    const float *Ws = (const float*)d_in[14], *bs = (const float*)d_in[15];
    const float *Wst= (const float*)d_in[16], *bst= (const float*)d_in[17];
    const float *Wo = (const float*)d_in[18], *bo = (const float*)d_in[19];

    const int Nn = in_sizes[0] / 5;        // 100000
    const int E  = in_sizes[1] / 2;        // 1200000

    float* out   = (float*)d_out;
    float* state = out;                    // [N,2]
    float* stab  = out + (size_t)Nn * 2;   // scalar
    float* opf   = stab + 1;               // scalar
    float* H3    = opf + 1;                // [N,64] final h, written in place

    float* ws   = (float*)d_ws;
    float* icnt = ws;                             // N
    float* xpad = icnt + Nn;                      // 8N  (x padded 5->8)
    float* xbar = xpad + (size_t)Nn * 8;          // 64N (layer1 uses 8N, stride 8)
    float* hbuf = xbar + (size_t)Nn * 64;         // 64N
    float* wsc  = hbuf + (size_t)Nn * 64;
    float *T1 = wsc,       *M1 = T1 + 8 * 64,  *d1 = M1 + 8 * 64;
    float *T2 = d1 + 64,   *M2 = T2 + 64 * 64, *d2 = M2 + 64 * 64;
    float *T3 = d2 + 64,   *M3 = T3 + 64 * 64, *d3 = M3 + 64 * 64;

    const int TPB = 256;
    int gN   = (Nn + TPB - 1) / TPB;
    int gE   = (E + TPB - 1) / TPB;
    int gE8  = (E * 8 + TPB - 1) / TPB;
    int g8N  = (Nn * 8 + TPB - 1) / TPB;
    int g64N = (Nn * 64 + TPB - 1) / TPB;
    int tiles = (Nn + 15) / 16;
    int gG   = (tiles + 7) / 8;

    // degree counts -> inv_cnt (self loop included), zero scalar accumulators
    pignn_init_cnt<<<gN, TPB, 0, stream>>>(icnt, Nn, stab, opf);
    pignn_count<<<gE, TPB, 0, stream>>>(ei, E, icnt);
    pignn_recip<<<gN, TPB, 0, stream>>>(icnt, Nn);

    // fold message linears into update linears (tiny, once); layer1 K padded to 8
    pignn_fuse_w<<<64, TPB, 0, stream>>>(W1, b1, U1, c1, 5,  8,  T1, M1, d1);
    pignn_fuse_w<<<64, TPB, 0, stream>>>(W2, b2, U2, c2, 64, 64, T2, M2, d2);
    pignn_fuse_w<<<64, TPB, 0, stream>>>(W3, b3, U3, c3, 64, 64, T3, M3, d3);

    // layer 1: pad x, aggregate 5-wide into stride-8 xbar, WMMA update -> hbuf
    pignn_pad5to8<<<g8N, TPB, 0, stream>>>(xpad, x, Nn);            // padded x
    pignn_pad5to8<<<g8N, TPB, 0, stream>>>(xbar, x, Nn);            // self loop init
    pignn_scatter5<<<gE, TPB, 0, stream>>>(ei, E, x, xbar);
    pignn_gemm<8><<<gG, TPB, 0, stream>>>(xpad, xbar, icnt, T1, M1, d1, hbuf, Nn);

    // layer 2: aggregate 64-wide, update in place (row-exclusive per wave)
    pignn_copy<<<g64N, TPB, 0, stream>>>(xbar, hbuf, Nn * 64);
    pignn_scatter64<<<gE8, TPB, 0, stream>>>(ei, E, hbuf, xbar);
    pignn_gemm<64><<<gG, TPB, 0, stream>>>(hbuf, xbar, icnt, T2, M2, d2, hbuf, Nn);

    // layer 3: write h straight into its output slot
    pignn_copy<<<g64N, TPB, 0, stream>>>(xbar, hbuf, Nn * 64);
    pignn_scatter64<<<gE8, TPB, 0, stream>>>(ei, E, hbuf, xbar);
    pignn_gemm<64><<<gG, TPB, 0, stream>>>(hbuf, xbar, icnt, T3, M3, d3, H3, Nn);

    // heads: state, mean(sigmoid(h@Wst+bst)), mean(h@Wo+bo)
    pignn_heads<<<gN, TPB, 0, stream>>>(H3, Ws, bs, Wst, bst, Wo, bo,
                                        state, stab, opf, Nn);
}